// DNN_KalmanNet_SLAM_38027640439430
// MI455X (gfx1250) — compile-verified
//
#include <hip/hip_runtime.h>
#include <math.h>

// ---------------------------------------------------------------------------
// KalmanNet single-step inference on MI455X (gfx1250, wave32).
// One workgroup (16 waves), all stages chained through LDS, matvecs done on
// the fp32 matrix pipe via V_WMMA_F32_16X16X4_F32 (exact fp32 math).
//
// Roofline: ~0.45 M MACs vs ~1.8 MB fp32 weights -> AI ~0.5 FLOP/byte,
// pure latency/bandwidth bound; whole model is L2-resident (192 MB L2).
// Single persistent workgroup avoids 7 kernel-launch round trips.
// ---------------------------------------------------------------------------

typedef __attribute__((ext_vector_type(2))) float v2f;
typedef __attribute__((ext_vector_type(8))) float v8f;

#define XD 5
#define YD 2
#define GRUH 145            // 5*(5*5+2*2)
#define H1 560
#define H2 40
#define NT 512              // 16 wave32s on one WGP

#if __has_builtin(__builtin_amdgcn_wmma_f32_16x16x4_f32)
#define HAVE_WMMA_F32X4 1
#endif

__device__ __forceinline__ float sigmoidf_(float x) {
    return 1.0f / (1.0f + expf(-x));
}

// One wave computes y[outBase .. outBase+15] = act(W(M,K) @ xs + bias).
// xs must be zero-padded in LDS up to ceil(K/4)*4 elements.
// Mapping: A = broadcast x-chunk (16x4), B = W^T tile (4x16), C accum.
// All 16 rows of D are identical; lane n<16 extracts D[0][n] from acc[0].
// Ragged tail: x zero-pad kills invalid columns, so W loads are merely
// address-clamped (min(col, K-1)) -> no exec divergence around WMMA.
__device__ __forceinline__ void wave_gemv16(
    const float* __restrict__ W, const float* __restrict__ bias,
    const float* __restrict__ xs, float* __restrict__ ys,
    int M, int K, int outBase, int lane, bool relu)
{
    const int n = lane & 15;
    int row = outBase + n;
    if (row >= M) row = M - 1;                 // clamp: computed but not stored
    const int k0 = (lane < 16) ? 0 : 2;        // which K-pair this lane holds
    const float* __restrict__ Wr = W + (long)row * K;

#ifdef HAVE_WMMA_F32X4
    v8f acc = {};
#else
    float partial = 0.0f;
#endif

    const int full = K >> 2;                   // chunks where all 4 k valid
#pragma unroll 4
    for (int kc = 0; kc < full; ++kc) {
        const int c0 = (kc << 2) + k0;
        v2f a, b;
        a.x = xs[c0];     a.y = xs[c0 + 1];
        b.x = Wr[c0];     b.y = Wr[c0 + 1];
#ifdef HAVE_WMMA_F32X4
        // (neg_a, A, neg_b, B, c_mod, C, reuse_a, reuse_b)
        acc = __builtin_amdgcn_wmma_f32_16x16x4_f32(
            false, a, false, b, (short)0, acc, false, false);
#else
        partial += a.x * b.x + a.y * b.y;
#endif
    }
    if (K & 3) {                               // ragged tail chunk
        const int c0 = (full << 2) + k0;
        const int ca = (c0     < K - 1) ? c0     : (K - 1);  // in-bounds clamp
        const int cb = (c0 + 1 < K - 1) ? c0 + 1 : (K - 1);
        v2f a, b;
        a.x = xs[c0];     a.y = xs[c0 + 1];    // zero-padded -> kills junk W
        b.x = Wr[ca];
        b.y = Wr[cb];
#ifdef HAVE_WMMA_F32X4
        acc = __builtin_amdgcn_wmma_f32_16x16x4_f32(
            false, a, false, b, (short)0, acc, false, false);
#else
        partial += a.x * b.x + a.y * b.y;
#endif
    }

#ifdef HAVE_WMMA_F32X4
    float result = acc[0];                     // D[0][n], full K-sum done by HW
#else
    partial += __shfl_xor(partial, 16, 32);    // fold K={2,3} half onto lanes<16
    float result = partial;
#endif

    if (lane < 16) {
        const int r = outBase + n;
        if (r < M) {
            float v = result + bias[r];
            if (relu) v = fmaxf(v, 0.0f);
            ys[r] = v;
        }
    }
}

__device__ __forceinline__ void gru_pointwise(
    const float* __restrict__ gi, const float* __restrict__ gh,
    const float* __restrict__ hprev, float* __restrict__ hnew, int tid)
{
    for (int j = tid; j < GRUH; j += NT) {
        float r = sigmoidf_(gi[j] + gh[j]);
        float z = sigmoidf_(gi[GRUH + j] + gh[GRUH + j]);
        float nn = tanhf(gi[2 * GRUH + j] + r * gh[2 * GRUH + j]);
        hnew[j] = (1.0f - z) * nn + z * hprev[j];
    }
}

__global__ __launch_bounds__(NT) void kalmannet_kernel(
    const float* __restrict__ state_inno, const float* __restrict__ obs_inno,
    const float* __restrict__ diff_state, const float* __restrict__ diff_obs,
    const float* __restrict__ W1,   const float* __restrict__ b1,
    const float* __restrict__ Wih0, const float* __restrict__ Whh0,
    const float* __restrict__ bih0, const float* __restrict__ bhh0,
    const float* __restrict__ Wih1, const float* __restrict__ Whh1,
    const float* __restrict__ bih1, const float* __restrict__ bhh1,
    const float* __restrict__ W2a,  const float* __restrict__ b2a,
    const float* __restrict__ W2b,  const float* __restrict__ b2b,
    const float* __restrict__ hn,   float* __restrict__ out)
{
    __shared__ float xbuf[16];        // 14 + zero pad
    __shared__ float l1out[H1];       // 560 (multiple of 4)
    __shared__ float gi[448];         // 435 rounded to tile
    __shared__ float gh[448];
    __shared__ float hprev0[160];     // 145 + zero pad to 37*4=148
    __shared__ float hprev1[160];
    __shared__ float h0new[160];
    __shared__ float h1new[160];
    __shared__ float l2h[48];         // 40
    __shared__ float obuf[16];

    const int tid   = threadIdx.x;
    const int lane  = tid & 31;
    const int wave  = tid >> 5;
    const int nwav  = NT >> 5;

    // Zero the buffers whose padding is consumed as A-side (x) input.
    if (tid < 16) xbuf[tid] = 0.0f;
    for (int i = tid; i < 160; i += NT) {
        hprev0[i] = 0.0f; hprev1[i] = 0.0f; h0new[i] = 0.0f; h1new[i] = 0.0f;
    }
    if (tid < 48) l2h[tid] = 0.0f;
    __syncthreads();

    // Stage 0: x = concat(state_inno, obs_inno, diff_state, diff_obs); load hn
    if (tid < XD)                xbuf[tid]          = state_inno[tid];
    else if (tid < XD + YD)      xbuf[tid]          = obs_inno[tid - XD];
    else if (tid < 2*XD + YD)    xbuf[tid]          = diff_state[tid - XD - YD];
    else if (tid < 2*XD + 2*YD)  xbuf[tid]          = diff_obs[tid - 2*XD - YD];
    for (int i = tid; i < GRUH; i += NT) {
        hprev0[i] = hn[i];
        hprev1[i] = hn[GRUH + i];
    }
    __syncthreads();

    // Stage 1: l1_out = relu(W1 @ x + b1)            (560x14 -> 35 tiles)
    for (int t = wave; t < 35; t += nwav)
        wave_gemv16(W1, b1, xbuf, l1out, H1, 2*XD + 2*YD, t * 16, lane, true);
    __syncthreads();

    // Stage 2: GRU0 gates: gi = Wih0@l1+bih0 (28 tiles, K=560)
    //                      gh = Whh0@h0+bhh0 (28 tiles, K=145)
    for (int t = wave; t < 56; t += nwav) {
        if (t < 28) wave_gemv16(Wih0, bih0, l1out,  gi, 3*GRUH, H1,   t * 16,       lane, false);
        else        wave_gemv16(Whh0, bhh0, hprev0, gh, 3*GRUH, GRUH, (t - 28) * 16, lane, false);
    }
    __syncthreads();
    gru_pointwise(gi, gh, hprev0, h0new, tid);
    __syncthreads();

    // Stage 3: GRU1 gates (both K=145)
    for (int t = wave; t < 56; t += nwav) {
        if (t < 28) wave_gemv16(Wih1, bih1, h0new,  gi, 3*GRUH, GRUH, t * 16,        lane, false);
        else        wave_gemv16(Whh1, bhh1, hprev1, gh, 3*GRUH, GRUH, (t - 28) * 16, lane, false);
    }
    __syncthreads();
    gru_pointwise(gi, gh, hprev1, h1new, tid);
    __syncthreads();

    // Stage 4: l2_hidden = relu(W2a @ h1 + b2a)      (40x145 -> 3 tiles)
    for (int t = wave; t < 3; t += nwav)
        wave_gemv16(W2a, b2a, h1new, l2h, H2, GRUH, t * 16, lane, true);
    __syncthreads();

    // Stage 5: out = W2b @ l2_hidden + b2b           (10x40 -> 1 tile)
    if (wave == 0)
        wave_gemv16(W2b, b2b, l2h, obuf, XD * YD, H2, 0, lane, false);
    __syncthreads();

    if (tid < XD * YD) out[tid] = obuf[tid];
}

extern "C" void kernel_launch(void* const* d_in, const int* in_sizes, int n_in,
                              void* d_out, int out_size, void* d_ws, size_t ws_size,
                              hipStream_t stream)
{
    (void)in_sizes; (void)n_in; (void)out_size; (void)d_ws; (void)ws_size;
    const float* p[19];
    for (int i = 0; i < 19; ++i) p[i] = (const float*)d_in[i];
    kalmannet_kernel<<<1, NT, 0, stream>>>(
        p[0], p[1], p[2], p[3],          // x pieces
        p[4], p[5],                      // W1, b1
        p[6], p[7], p[8], p[9],          // GRU0
        p[10], p[11], p[12], p[13],      // GRU1
        p[14], p[15], p[16], p[17],      // l2
        p[18],                           // hn
        (float*)d_out);
}